// AdaptiveNodeCollapse_12309376270852
// MI455X (gfx1250) — compile-verified
//
#include <hip/hip_runtime.h>
#include <hip/hip_bf16.h>
#include <math.h>

typedef __attribute__((ext_vector_type(2))) float v2f;
typedef __attribute__((ext_vector_type(8))) float v8f;

#define BATCH 64
#define QN    256
#define DN    768
#define THRESH 0.9f
#define EPSN   1e-12f

// ---------------------------------------------------------------------------
// K0: per-row L2 norms (clamped to eps) + zero the 256x256 flag bitmask.
// One wave32 per row; butterfly reduction.
// ---------------------------------------------------------------------------
__global__ void k_norms(const float* __restrict__ C,
                        float* __restrict__ norms,
                        unsigned* __restrict__ flags) {
    int tid  = threadIdx.x;
    int wave = tid >> 5;
    int lane = tid & 31;
    if ((int)blockIdx.x < 8) flags[blockIdx.x * 256 + tid] = 0u;  // 2048 words
    int row = (int)blockIdx.x * 8 + wave;                          // 0..16383
    if (row >= BATCH * QN) return;
    const float* p = C + (size_t)row * DN;
    float s = 0.f;
    for (int k = lane; k < DN; k += 32) { float v = p[k]; s += v * v; }
    #pragma unroll
    for (int off = 16; off; off >>= 1) s += __shfl_xor(s, off, 32);
    if (lane == 0) norms[row] = fmaxf(sqrtf(s), EPSN);
}

// ---------------------------------------------------------------------------
// K1: gram + threshold.  One wave per (upper-tri 16x16 tile pair, batch).
// sim tile accumulated with V_WMMA_F32_16X16X4_F32 over D=768 (192 steps).
// flag bit set iff dot > 0.9 * n_i * n_j and i < j (OR across batches).
// ---------------------------------------------------------------------------
__global__ void k_gram(const float* __restrict__ C,
                       const float* __restrict__ norms,
                       unsigned* __restrict__ flags) {
    int lane = threadIdx.x & 31;
    int wave = threadIdx.x >> 5;
    int p = (int)blockIdx.x * 8 + wave;          // tile-pair index, 0..135
    if (p >= 136) return;                         // wave-uniform guard
    int b = blockIdx.y;
    int ti = 0;
    while (p >= 16 - ti) { p -= 16 - ti; ++ti; }
    int tj = ti + p;                              // ti <= tj

    int h = lane >> 4, l = lane & 15;
    // A[m][k] fragment: lane (m + 16h) holds K = k+2h, k+2h+1  (ISA 16x4 f32 layout)
    const float* Ap = C + ((size_t)b * QN + ti * 16 + l) * DN + 2 * h;
    const float* Bp = C + ((size_t)b * QN + tj * 16 + l) * DN + 2 * h;

    v8f acc = {};
    for (int k = 0; k < DN; k += 4) {
        v2f a  = *(const v2f*)(Ap + k);
        v2f bv = *(const v2f*)(Bp + k);
        acc = __builtin_amdgcn_wmma_f32_16x16x4_f32(
                  false, a, false, bv, (short)0, acc, false, false);
    }

    int j = tj * 16 + l;                          // column index (per lane)
    float nj = norms[b * QN + j];
    #pragma unroll
    for (int r = 0; r < 8; ++r) {
        int i = ti * 16 + r + 8 * h;              // C/D layout: M = r + 8h, N = l
        if (i < j) {
            float ni = norms[b * QN + i];
            if (acc[r] > THRESH * ni * nj)
                atomicOr(&flags[i * 8 + (j >> 5)], 1u << (j & 31));
        }
    }
}

// ---------------------------------------------------------------------------
// K2: build merge matrix M (256x256) in LDS.  The sequential pair-scan is a
// composition of linear row ops; thread t owns column t exclusively, so the
// whole ordered scan needs NO barriers.  __ffs walks set bits j-ascending,
// preserving the reference's row-major pair order.
// ---------------------------------------------------------------------------
__global__ void k_buildM(const unsigned* __restrict__ flags,
                         float* __restrict__ Mg) {
    extern __shared__ float Ml[];                 // 256*256 floats = 256 KB LDS
    int t = threadIdx.x;
    for (int r = 0; r < QN; ++r) Ml[r * QN + t] = (r == t) ? 1.f : 0.f;

    for (int i = 0; i < QN - 1; ++i) {
        float vi = Ml[i * QN + t];
        bool dirty = false;
        for (int w = (i + 1) >> 5; w < 8; ++w) {  // flags are strict-upper by construction
            unsigned bits = flags[i * 8 + w];
            while (bits) {
                int j = (w << 5) + __ffs(bits) - 1;   // lowest bit first => j ascending
                bits &= bits - 1;
                float a = 0.5f * (vi + Ml[j * QN + t]);
                vi = a;
                Ml[j * QN + t] = a;
                dirty = true;
            }
        }
        if (dirty) Ml[i * QN + t] = vi;
    }
    for (int r = 0; r < QN; ++r) Mg[r * QN + t] = Ml[r * QN + t];
}

// ---------------------------------------------------------------------------
// K3: out[b] = M @ C[b]  (256x256 @ 256x768) via WMMA f32 16x16x4.
// One wave per 16x64 output strip: A fragment reused across 4 accumulators.
// ---------------------------------------------------------------------------
__global__ void k_apply(const float* __restrict__ C,
                        const float* __restrict__ Mg,
                        float* __restrict__ out) {
    int lane = threadIdx.x & 31;
    int h = lane >> 4, l = lane & 15;
    int qtile  = (int)blockIdx.x & 15;            // 16 q-tiles
    int dstrip = (int)blockIdx.x >> 4;            // 12 strips of 64 cols
    int b = blockIdx.y;
    int q0 = qtile * 16, d0 = dstrip * 64;

    const float* Arow = Mg + (size_t)(q0 + l) * QN + 2 * h;
    const float* Cb   = C + (size_t)b * QN * DN;

    v8f acc0 = {}, acc1 = {}, acc2 = {}, acc3 = {};
    for (int k = 0; k < QN; k += 4) {
        v2f a = *(const v2f*)(Arow + k);
        const float* Bp = Cb + (size_t)(k + 2 * h) * DN + d0 + l;
        v2f b0, b1, b2, b3;
        b0.x = Bp[0];  b0.y = Bp[DN];
        b1.x = Bp[16]; b1.y = Bp[DN + 16];
        b2.x = Bp[32]; b2.y = Bp[DN + 32];
        b3.x = Bp[48]; b3.y = Bp[DN + 48];
        acc0 = __builtin_amdgcn_wmma_f32_16x16x4_f32(false, a, false, b0, (short)0, acc0, false, false);
        acc1 = __builtin_amdgcn_wmma_f32_16x16x4_f32(false, a, false, b1, (short)0, acc1, false, false);
        acc2 = __builtin_amdgcn_wmma_f32_16x16x4_f32(false, a, false, b2, (short)0, acc2, false, false);
        acc3 = __builtin_amdgcn_wmma_f32_16x16x4_f32(false, a, false, b3, (short)0, acc3, false, false);
    }

    float* O = out + (size_t)b * QN * DN;
    #pragma unroll
    for (int r = 0; r < 8; ++r) {
        float* orow = O + (size_t)(q0 + r + 8 * h) * DN + d0 + l;
        orow[0]  = acc0[r];
        orow[16] = acc1[r];
        orow[32] = acc2[r];
        orow[48] = acc3[r];
    }
}

// ---------------------------------------------------------------------------
extern "C" void kernel_launch(void* const* d_in, const int* in_sizes, int n_in,
                              void* d_out, int out_size, void* d_ws, size_t ws_size,
                              hipStream_t stream) {
    const float* C = (const float*)d_in[0];       // [64, 256, 768] fp32
    float* out = (float*)d_out;                   // [64, 256, 768] fp32
    char* ws = (char*)d_ws;

    float*    norms = (float*)ws;                 // 16384 floats  = 65536 B
    unsigned* flags = (unsigned*)(ws + 65536);    // 2048 u32      =  8192 B
    float*    Mg    = (float*)(ws + 73728);       // 65536 floats  = 262144 B

    // K0: norms + flag clear
    k_norms<<<dim3(2048), dim3(256), 0, stream>>>(C, norms, flags);
    // K1: gram + threshold -> flag bitmask (136 tile-pairs x 64 batches, 8 waves/block)
    k_gram<<<dim3(17, 64), dim3(256), 0, stream>>>(C, norms, flags);
    // K2: build 256x256 merge matrix in 256KB LDS (single workgroup, no barriers)
    k_buildM<<<dim3(1), dim3(256), QN * QN * sizeof(float), stream>>>(flags, Mg);
    // K3: out = M @ C per batch (16 q-tiles x 12 d-strips, 64 batches, wave32 blocks)
    k_apply<<<dim3(16 * 12, 64), dim3(32), 0, stream>>>(C, Mg, out);
}